// SO3Projector_13494787244424
// MI455X (gfx1250) — compile-verified
//
#include <hip/hip_runtime.h>
#include <stdint.h>

// ---------------------------------------------------------------------------
// SO3 projector: per-matrix nearest-rotation (polar factor of sign(det A)*A^T)
// Memory-bound streaming kernel (128 B/matrix, ~5.5 us HBM floor on MI455X).
// Data movement uses CDNA5-specific paths:
//   * TDM tensor_load_to_lds (one DMA per block, HW-padded 72B LDS stride)
//   * global_store_async_from_lds_b64 (ASYNCcnt) for the output
// ---------------------------------------------------------------------------

#define TPB 256
#define STRIDE_F 18u                 // 72-byte padded LDS stride per matrix
#define LDS_FLOATS (TPB * STRIDE_F)  // 18432 B

typedef uint32_t u32x4 __attribute__((ext_vector_type(4)));
typedef uint32_t u32x8 __attribute__((ext_vector_type(8)));

union G0u { u32x4 v; uint32_t u[4]; };
union G1u { u32x8 v; uint32_t u[8]; };

// 32-bit LDS byte offset of a generic pointer that points into LDS.
__device__ __forceinline__ uint32_t lds_byte_off(void* p) {
  return (uint32_t)(uintptr_t)(__attribute__((address_space(3))) void*)p;
}

__global__ void __launch_bounds__(TPB)
so3_polar_kernel(const float* __restrict__ in, float* __restrict__ out, int nmat) {
  __shared__ __align__(16) float smat[LDS_FLOATS];

  const int t         = threadIdx.x;
  const int blockMat0 = blockIdx.x * TPB;
  const int matsHere  = min(TPB, nmat - blockMat0);
  const float* gsrc   = in  + (size_t)blockMat0 * 16;
  float*       gdst   = out + (size_t)blockMat0 * 16;
  const uint32_t lbase = lds_byte_off((void*)smat);

  // ---- Phase 1: TDM bulk DMA global -> LDS (wave 0 issues once per block).
  // D# pad: after every 16 dwords (one 64B matrix) skip 2 dwords -> 72B stride.
  if (t < 32) {
    const uint64_t ga   = (uint64_t)(uintptr_t)gsrc;
    const uint32_t dim0 = (uint32_t)matsHere * 16u;   // elements of 4 bytes

    G0u g0;
    g0.u[0] = 1u;                                     // count=1 (valid user D#)
    g0.u[1] = lbase;                                  // lds_addr (bytes)
    g0.u[2] = (uint32_t)(ga & 0xFFFFFFFFu);           // global_addr[31:0]
    g0.u[3] = (uint32_t)((ga >> 32) & 0x01FFFFFFu)    // global_addr[56:32]
            | 0x80000000u;                            // type=2 ("image")

    G1u g1;
    g1.u[0] = (2u << 16)     // data_size = 4 bytes
            | (1u << 20)     // pad_enable
            | (3u << 22)     // pad_interval: 16 dwords (64 B)
            | (1u << 25);    // pad_amount:   2 dwords (8 B)
    g1.u[1] = (dim0 & 0xFFFFu) << 16;                 // tensor_dim0[15:0]
    g1.u[2] = (dim0 >> 16) | (1u << 16);              // tensor_dim0[31:16] | tensor_dim1=1
    g1.u[3] = (dim0 & 0xFFFFu) << 16;                 // tile_dim0 (<=4096 fits 16b)
    g1.u[4] = 0u;                                     // tile_dim1/2 unused (1D)
    g1.u[5] = dim0;                                   // tensor_dim0_stride (unused 1D)
    g1.u[6] = 0u;
    g1.u[7] = 0u;

    asm volatile("tensor_load_to_lds %0, %1" :: "s"(g0.v), "s"(g1.v) : "memory");
#if __has_builtin(__builtin_amdgcn_s_wait_tensorcnt)
    __builtin_amdgcn_s_wait_tensorcnt(0);
#else
    asm volatile("s_wait_tensorcnt 0x0" ::: "memory");
#endif
  }
  __syncthreads();

  // ---- Phase 2: per-thread polar decomposition (bank-conflict-free LDS reads)
  if (t < matsHere) {
    const float* mp = &smat[(uint32_t)t * STRIDE_F];
    float a00 = mp[0], a01 = mp[1], a02 = mp[2],  p0 = mp[3];
    float a10 = mp[4], a11 = mp[5], a12 = mp[6],  p1 = mp[7];
    float a20 = mp[8], a21 = mp[9], a22 = mp[10], p2 = mp[11];

    float m00 = a11*a22 - a12*a21;
    float m01 = a12*a20 - a10*a22;
    float m02 = a10*a21 - a11*a20;
    float det = a00*m00 + a01*m01 + a02*m02;
    float sgn = (det < 0.0f) ? -1.0f : 1.0f;
    float fro2 = a00*a00+a01*a01+a02*a02 + a10*a10+a11*a11+a12*a12
               + a20*a20+a21*a21+a22*a22;

    // X = sgn * A^T  (det X = |det A| >= 0). Rb = polar(X) = sgn * V U^T.
    float x00 = sgn*a00, x01 = sgn*a10, x02 = sgn*a20;
    float x10 = sgn*a01, x11 = sgn*a11, x12 = sgn*a21;
    float x20 = sgn*a02, x21 = sgn*a12, x22 = sgn*a22;

    // Scaled Newton polar iteration: X <- 0.5*(g*X + (1/(g*d))*cof(X)),
    // g = sqrt(||X^-1||_F / ||X||_F). Quadratic convergence; 10 iters >> fp32.
    #pragma unroll
    for (int it = 0; it < 10; ++it) {
      float c00 = x11*x22 - x12*x21, c01 = x12*x20 - x10*x22, c02 = x10*x21 - x11*x20;
      float c10 = x02*x21 - x01*x22, c11 = x00*x22 - x02*x20, c12 = x01*x20 - x00*x21;
      float c20 = x01*x12 - x02*x11, c21 = x02*x10 - x00*x12, c22 = x00*x11 - x01*x10;
      float d  = x00*c00 + x01*c01 + x02*c02;
      float nX = x00*x00+x01*x01+x02*x02 + x10*x10+x11*x11+x12*x12
               + x20*x20+x21*x21+x22*x22;
      float nC = c00*c00+c01*c01+c02*c02 + c10*c10+c11*c11+c12*c12
               + c20*c20+c21*c21+c22*c22;
      float g  = sqrtf(sqrtf(nC / nX) / fabsf(d));
      float u  = 0.5f * g;
      float w  = 0.5f / (g * d);
      x00 = u*x00 + w*c00; x01 = u*x01 + w*c01; x02 = u*x02 + w*c02;
      x10 = u*x10 + w*c10; x11 = u*x11 + w*c11; x12 = u*x12 + w*c12;
      x20 = u*x20 + w*c20; x21 = u*x21 + w*c21; x22 = u*x22 + w*c22;
    }

    // trans = -Rb * (sqrt3 * sgn * t / fro); fold the minus into k.
    const float k = -1.7320508075688772f * sgn * rsqrtf(fro2);
    float s0 = k*p0, s1 = k*p1, s2 = k*p2;
    float tr0 = x00*s0 + x01*s1 + x02*s2;
    float tr1 = x10*s0 + x11*s1 + x12*s2;
    float tr2 = x20*s0 + x21*s1 + x22*s2;

    float* op = &smat[(uint32_t)t * STRIDE_F];   // same disjoint region, no race
    op[0]  = x00; op[1]  = x01; op[2]  = x02; op[3]  = tr0;
    op[4]  = x10; op[5]  = x11; op[6]  = x12; op[7]  = tr1;
    op[8]  = x20; op[9]  = x21; op[10] = x22; op[11] = tr2;
    op[12] = 0.0f; op[13] = 0.0f; op[14] = 0.0f; op[15] = 1.0f;
  }
  __syncthreads();

  // ---- Phase 3: async store LDS -> global (fully coalesced 8B chunks,
  // de-padding the 72B LDS stride back to the dense 64B output layout).
  const uint32_t nchunk = (uint32_t)matsHere * 8u;   // 8 x 8B per matrix
  #pragma unroll
  for (int r = 0; r < 8; ++r) {
    uint32_t c = (uint32_t)(r * TPB) + (uint32_t)t;
    if (c < nchunk) {
      uint32_t m = c >> 3, q = c & 7u;
      uint32_t laddr = lbase + m * 72u + q * 8u;    // padded LDS source
      uint32_t goff  = c * 8u;                      // dense global dest offset
      asm volatile("global_store_async_from_lds_b64 %0, %1, %2"
                   :: "v"(goff), "v"(laddr), "s"(gdst) : "memory");
    }
  }
  asm volatile("s_wait_asynccnt 0x0" ::: "memory");
}

extern "C" void kernel_launch(void* const* d_in, const int* in_sizes, int n_in,
                              void* d_out, int out_size, void* d_ws, size_t ws_size,
                              hipStream_t stream) {
  (void)n_in; (void)out_size; (void)d_ws; (void)ws_size;
  const float* in  = (const float*)d_in[0];
  float*       out = (float*)d_out;
  const int nmat = in_sizes[0] / 16;                // 524288 4x4 matrices
  const int grid = (nmat + TPB - 1) / TPB;          // 2048 blocks of 256 (8 waves)
  so3_polar_kernel<<<grid, TPB, 0, stream>>>(in, out, nmat);
}